// TopKGating_3367254360369
// MI455X (gfx1250) — compile-verified
//
#include <hip/hip_runtime.h>
#include <math.h>

typedef __attribute__((ext_vector_type(2))) float v2f;
typedef __attribute__((ext_vector_type(4))) float v4f;
typedef __attribute__((ext_vector_type(8))) float v8f;

#define D_IN   1024
#define NE     64
#define ROWS   128     // tokens per block
#define BK     32      // K-chunk staged through LDS
#define APAD   36      // x-tile row stride (floats): 16B aligned, conflict-free frags
#define EPAD   68      // logits row stride (floats): 8B aligned, conflict-free frags
#define ALPHA_ 10.0f
#define EPS_   1e-5f

// Pack gate_w [E=64][D=1024] into WMMA B-fragment order so the hot loop's B
// loads are contiguous:  bp[((ktg*4+et)*32+lane)*2+j] = B[k][e] = gate_w[e][k]
// with k = ktg*4 + ((lane>>4)<<1) + j, e = et*16 + (lane&15).
__global__ __launch_bounds__(256) void pack_b_kernel(const float* __restrict__ gw,
                                                     float* __restrict__ bp) {
  int gid = blockIdx.x * 256 + threadIdx.x;     // 0..65535
  int j   = gid & 1;
  int l   = (gid >> 1) & 31;
  int et  = (gid >> 6) & 3;
  int ktg = gid >> 8;
  int k = ktg * 4 + ((l >> 4) << 1) + j;
  int e = et * 16 + (l & 15);
  bp[gid] = gw[e * D_IN + k];
}

__global__ __launch_bounds__(256) void topk_gating_kernel(
    const float* __restrict__ x, const float* __restrict__ bp,
    const float* __restrict__ gate_b, const float* __restrict__ w_noise,
    const float* __restrict__ noise, float* __restrict__ out) {
  __shared__ float smem[ROWS * EPAD];        // 34,816 B
  float* a_s = smem;                          // phase 1: 128*36 = 4608 floats
  float* b_s = smem + ROWS * APAD;            // phase 1: 2048 floats
  float* l_s = smem;                          // phase 2/3: 128*68 floats

  const int tid  = threadIdx.x;
  const int lane = tid & 31;
  const int wave = tid >> 5;
  const int lm   = lane & 15;                 // M (A) / N (B,C) index within frag
  const int koff = (lane >> 4) << 1;          // K offset 0 or 2 for A/B frags
  const int wrow = wave * 16;                 // wave's token rows within block
  const int row0 = blockIdx.x * ROWS;

  v8f c0 = {}, c1 = {}, c2 = {}, c3 = {};     // clean logits, 16 tok x 64 exp

  // ---------------- Phase 1: clean = x @ gate_w^T  (fp32 WMMA) -------------
  for (int kt = 0; kt < D_IN / BK; ++kt) {
    __syncthreads();
    // stage x tile [128 rows][32 k] -> LDS, coalesced b128 loads/stores
    #pragma unroll
    for (int i = 0; i < 4; ++i) {
      int idx = tid + i * 256;                // 1024 float4s
      int r = idx >> 3, c = (idx & 7) << 2;
      const float* src = x + (size_t)(row0 + r) * D_IN + kt * BK + c;
      *(v4f*)(a_s + r * APAD + c) = *(const v4f*)src;
      if (kt + 1 < D_IN / BK) __builtin_prefetch(src + BK, 0, 0);
    }
    // stage pre-packed B tile (2048 floats, L2-resident) -> LDS
    #pragma unroll
    for (int i = 0; i < 2; ++i) {
      int idx = (tid + i * 256) << 2;
      *(v4f*)(b_s + idx) = *(const v4f*)(bp + kt * 2048 + idx);
    }
    __syncthreads();
    #pragma unroll
    for (int k2 = 0; k2 < BK; k2 += 4) {
      v2f a = *(const v2f*)(a_s + (wrow + lm) * APAD + k2 + koff);
      const float* bb = b_s + (k2 >> 2) * 256 + lane * 2;
      v2f b0 = *(const v2f*)(bb);
      v2f b1 = *(const v2f*)(bb + 64);
      v2f b2 = *(const v2f*)(bb + 128);
      v2f b3 = *(const v2f*)(bb + 192);
      c0 = __builtin_amdgcn_wmma_f32_16x16x4_f32(false, a, false, b0, (short)0, c0, false, false);
      c1 = __builtin_amdgcn_wmma_f32_16x16x4_f32(false, a, false, b1, (short)0, c1, false, false);
      c2 = __builtin_amdgcn_wmma_f32_16x16x4_f32(false, a, false, b2, (short)0, c2, false, false);
      c3 = __builtin_amdgcn_wmma_f32_16x16x4_f32(false, a, false, b3, (short)0, c3, false, false);
    }
  }
  __syncthreads();

  // dump clean (+bias) to LDS so phase 2 can re-fragment it as a WMMA A matrix
  float gb0 = gate_b[lm], gb1 = gate_b[16 + lm], gb2 = gate_b[32 + lm], gb3 = gate_b[48 + lm];
  const int mb = wrow + ((lane >> 4) << 3);   // C layout: lanes 16-31 hold M+8
  #pragma unroll
  for (int v = 0; v < 8; ++v) {
    l_s[(mb + v) * EPAD +      lm] = c0[v] + gb0;
    l_s[(mb + v) * EPAD + 16 + lm] = c1[v] + gb1;
    l_s[(mb + v) * EPAD + 32 + lm] = c2[v] + gb2;
    l_s[(mb + v) * EPAD + 48 + lm] = c3[v] + gb3;
  }
  __syncthreads();

  // ---------------- Phase 2: raw = clean @ w_noise  (fp32 WMMA) ------------
  // w_noise is [64 k][64 e] row-major == already B-oriented; 16 KB, L2-hot.
  v8f r0 = {}, r1 = {}, r2 = {}, r3 = {};
  #pragma unroll
  for (int k2 = 0; k2 < NE; k2 += 4) {
    v2f a = *(const v2f*)(l_s + (wrow + lm) * EPAD + k2 + koff);
    const float* wn = w_noise + (k2 + koff) * NE + lm;
    v2f b0 = { wn[0],  wn[NE + 0]  };
    v2f b1 = { wn[16], wn[NE + 16] };
    v2f b2 = { wn[32], wn[NE + 32] };
    v2f b3 = { wn[48], wn[NE + 48] };
    r0 = __builtin_amdgcn_wmma_f32_16x16x4_f32(false, a, false, b0, (short)0, r0, false, false);
    r1 = __builtin_amdgcn_wmma_f32_16x16x4_f32(false, a, false, b1, (short)0, r1, false, false);
    r2 = __builtin_amdgcn_wmma_f32_16x16x4_f32(false, a, false, b2, (short)0, r2, false, false);
    r3 = __builtin_amdgcn_wmma_f32_16x16x4_f32(false, a, false, b3, (short)0, r3, false, false);
  }
  __syncthreads();   // all clean_s fragment reads complete before overwrite

  // noisy logits in C-fragment coords: logit = clean + noise*(softplus(raw)+eps)
  #pragma unroll
  for (int v = 0; v < 8; ++v) {
    const float* nz = noise + (size_t)(row0 + mb + v) * NE + lm;
    float rv, sp;
    rv = r0[v]; sp = rv > 0.f ? rv + log1pf(expf(-rv)) : log1pf(expf(rv));
    l_s[(mb + v) * EPAD +      lm] = (c0[v] + gb0) + nz[0]  * (sp + EPS_);
    rv = r1[v]; sp = rv > 0.f ? rv + log1pf(expf(-rv)) : log1pf(expf(rv));
    l_s[(mb + v) * EPAD + 16 + lm] = (c1[v] + gb1) + nz[16] * (sp + EPS_);
    rv = r2[v]; sp = rv > 0.f ? rv + log1pf(expf(-rv)) : log1pf(expf(rv));
    l_s[(mb + v) * EPAD + 32 + lm] = (c2[v] + gb2) + nz[32] * (sp + EPS_);
    rv = r3[v]; sp = rv > 0.f ? rv + log1pf(expf(-rv)) : log1pf(expf(rv));
    l_s[(mb + v) * EPAD + 48 + lm] = (c3[v] + gb3) + nz[48] * (sp + EPS_);
  }
  __syncthreads();

  // ---------------- Phase 3: per-token top-2 + double softmax --------------
  if (tid < ROWS) {
    float lg[NE];
    float m1 = -3.0e38f, m2 = -3.0e38f;
    #pragma unroll
    for (int e = 0; e < NE; ++e) {
      float l = l_s[tid * EPAD + e];
      lg[e] = l;
      if (l > m1) { m2 = m1; m1 = l; } else if (l > m2) { m2 = l; }
    }
    float sum = 0.f;
    #pragma unroll
    for (int e = 0; e < NE; ++e) { float s = expf(lg[e] - m1); lg[e] = s; sum += s; }
    const float thr = expf(m2 - m1);          // logit < m2  <=>  exp(l-m1) < thr
    const float rs = 1.f / sum;
    float omax = -3.0e38f;
    #pragma unroll
    for (int e = 0; e < NE; ++e) {
      float sm = lg[e] * rs;
      float o = (lg[e] < thr) ? ALPHA_ * log1pf(sm) : ALPHA_ * expm1f(sm);
      lg[e] = o;
      omax = fmaxf(omax, o);
    }
    float osum = 0.f;
    #pragma unroll
    for (int e = 0; e < NE; ++e) { float t = expf(lg[e] - omax); lg[e] = t; osum += t; }
    const float ro = 1.f / osum;
    float* op = out + (size_t)(row0 + tid) * NE;
    #pragma unroll
    for (int e = 0; e < NE; e += 4) {
      v4f vv = { lg[e] * ro, lg[e + 1] * ro, lg[e + 2] * ro, lg[e + 3] * ro };
      *(v4f*)(op + e) = vv;
    }
  }
}

extern "C" void kernel_launch(void* const* d_in, const int* in_sizes, int n_in,
                              void* d_out, int out_size, void* d_ws, size_t ws_size,
                              hipStream_t stream) {
  const float* x       = (const float*)d_in[0];   // [N, 1024]
  const float* gate_w  = (const float*)d_in[1];   // [64, 1024]
  const float* gate_b  = (const float*)d_in[2];   // [64]
  const float* w_noise = (const float*)d_in[3];   // [64, 64]
  const float* noise   = (const float*)d_in[4];   // [N, 64]
  float* out   = (float*)d_out;                   // [N, 64]
  float* bpack = (float*)d_ws;                    // 65536 floats = 256 KB

  const int n = in_sizes[0] / D_IN;               // 262144 tokens

  pack_b_kernel<<<256, 256, 0, stream>>>(gate_w, bpack);
  topk_gating_kernel<<<n / ROWS, 256, 0, stream>>>(x, bpack, gate_b, w_noise,
                                                   noise, out);
}